// EquivairantMultiheadAttention_4380866642599
// MI455X (gfx1250) — compile-verified
//
#include <hip/hip_runtime.h>
#include <cfloat>

// Problem constants (match reference)
#define B_    2
#define N_    2048
#define M_    64
#define C_    256
#define H_    8
#define POS_  6
#define D_    32
#define ROWS_ (B_ * N_)                 // 4096
#define SCALE_ 0.17677669529663687f     // 1/sqrt(32)
#define KST_  257                       // padded LDS row stride (dwords) for K tile

typedef float v2f __attribute__((ext_vector_type(2)));
typedef float v8f __attribute__((ext_vector_type(8)));

// TDM descriptor group vector types (ROCm 7.2 probe-confirmed arity: 5 args)
typedef unsigned int tdm_u32x4 __attribute__((ext_vector_type(4)));
typedef int          tdm_i32x4 __attribute__((ext_vector_type(4)));
typedef int          tdm_i32x8 __attribute__((ext_vector_type(8)));

#if defined(__has_builtin)
#if __has_builtin(__builtin_amdgcn_tensor_load_to_lds) && \
    __has_builtin(__builtin_amdgcn_s_wait_tensorcnt)
#define HAVE_TDM 1
#warning "CDNA5: TDM gather path ENABLED (tensor_load_to_lds + s_wait_tensorcnt)"
#endif
#endif
#ifndef HAVE_TDM
#define HAVE_TDM 0
#warning "CDNA5: TDM builtins NOT available - using manual LDS gather fallback"
#endif

__device__ __forceinline__ unsigned long long shfl_xor_u64(unsigned long long x,
                                                           int off) {
  unsigned lo = (unsigned)x, hi = (unsigned)(x >> 32);
  lo = __shfl_xor(lo, off, 32);
  hi = __shfl_xor(hi, off, 32);
  return ((unsigned long long)hi << 32) | lo;
}

// ---------------------------------------------------------------------------
// Kernel 1: top-M nearest neighbors per (b,n).
// Streams pairwise_g ONCE (201 MB) with non-temporal hints so the stream
// doesn't evict the L2-resident Q/K/V working set. Candidate keys
// (squared-dist bits << 32 | index) live entirely in REGISTERS (8/thread);
// each of the 64 selection rounds is: 7 register mins -> 5-step wave32
// shfl_xor reduction -> 8-slot LDS cross-wave combine with ONE barrier
// (double-buffered slots). Packed keys reproduce jax.lax.top_k ordering and
// lowest-index tie-breaking exactly; sqrt is monotonic so squared dists are
// order-equivalent (mask sentinel 1e8 -> 1e16).
// ---------------------------------------------------------------------------
__global__ void __launch_bounds__(256) topk_kernel(
    const float* __restrict__ pg,            // (B,N,N,POS)
    const unsigned char* __restrict__ mask,  // (B,N) bool
    int* __restrict__ nbhd_idx)              // (B*N, M)
{
  __shared__ unsigned long long wmin[2][8];
  const int row  = blockIdx.x;     // 0..B*N-1
  const int b    = row / N_;
  const int t    = threadIdx.x;
  const int lane = t & 31;
  const int wv   = t >> 5;
  const v2f* prow2 =
      reinterpret_cast<const v2f*>(pg + (size_t)row * N_ * POS_);

  unsigned long long key[8];
#pragma unroll
  for (int r = 0; r < 8; ++r) {
    const int j = t + 256 * r;                 // coalesced within wave
    v2f p0 = __builtin_nontemporal_load(prow2 + j * 3 + 0);
    v2f p1 = __builtin_nontemporal_load(prow2 + j * 3 + 1);
    v2f p2 = __builtin_nontemporal_load(prow2 + j * 3 + 2);
    float s = p0.x * p0.x + p0.y * p0.y + p1.x * p1.x + p1.y * p1.y +
              p2.x * p2.x + p2.y * p2.y;
    if (!mask[b * N_ + j]) s = 1e16f;
    key[r] = ((unsigned long long)__float_as_uint(s) << 32) | (unsigned)j;
  }

  for (int it = 0; it < M_; ++it) {
    unsigned long long kmin = key[0];
#pragma unroll
    for (int r = 1; r < 8; ++r) kmin = key[r] < kmin ? key[r] : kmin;
#pragma unroll
    for (int off = 16; off > 0; off >>= 1) {
      unsigned long long o = shfl_xor_u64(kmin, off);
      kmin = o < kmin ? o : kmin;
    }
    if (lane == 0) wmin[it & 1][wv] = kmin;
    __syncthreads();
    unsigned long long g = wmin[it & 1][0];
#pragma unroll
    for (int w = 1; w < 8; ++w) {
      unsigned long long o = wmin[it & 1][w];
      g = o < g ? o : g;
    }
    if (t == 0) nbhd_idx[row * M_ + it] = (int)(g & 0xFFFFFFFFu);
    // unique owner (index is in the key) removes the winner from its set
#pragma unroll
    for (int r = 0; r < 8; ++r)
      if (key[r] == g) key[r] = ~0ull;
  }
}

// ---------------------------------------------------------------------------
// Kernel 2: Y = X @ W^T + bias via V_WMMA_F32_16X16X4_F32 (fp32, exact).
// One wave per 16x16 output tile; compiler pipelines ~16 WMMAs with clause'd
// b64 fragment loads (verified in disasm) -- keep as is.
// ---------------------------------------------------------------------------
__global__ void __launch_bounds__(256) gemm_xwt_kernel(
    const float* __restrict__ X,     // (R, 256) row-major
    const float* __restrict__ W,     // (256, 256) row-major; Y = X * W^T
    const float* __restrict__ bias,  // (256)
    float* __restrict__ Y,           // (R, 256)
    int R)
{
  const int lane = threadIdx.x & 31;
  const int wave = threadIdx.x >> 5;
  const int tile = blockIdx.x * 8 + wave;   // global 16x16 tile id
  const int tilesPerRow = C_ / 16;          // 16 column tiles
  const int tr = tile / tilesPerRow;
  const int tc = tile % tilesPerRow;
  if (tr * 16 >= R) return;                 // wave-uniform

  const int mrow  = tr * 16 + (lane & 15);
  const int ncol  = tc * 16 + (lane & 15);
  const int khalf = (lane >> 4) * 2;        // 0 or 2
  const float* xrow = X + (size_t)mrow * C_ + khalf;
  const float* wrow = W + (size_t)ncol * C_ + khalf;

  v8f acc = {};
  for (int k0 = 0; k0 < C_; k0 += 4) {
    v2f a, bm;
    a.x  = xrow[k0];  a.y  = xrow[k0 + 1];
    bm.x = wrow[k0];  bm.y = wrow[k0 + 1];
    acc = __builtin_amdgcn_wmma_f32_16x16x4_f32(
        /*neg_a=*/false, a, /*neg_b=*/false, bm,
        /*c_mod=*/(short)0, acc, /*reuse_a=*/false, /*reuse_b=*/false);
  }

  const float bv   = bias[ncol];
  const int  rbase = tr * 16 + ((lane >> 4) << 3);
#pragma unroll
  for (int r = 0; r < 8; ++r)
    Y[(size_t)(rbase + r) * C_ + ncol] = acc[r] + bv;
}

// ---------------------------------------------------------------------------
// Kernel 3: fused relative-position attention per (b,n) row.
// The 64 gathered K rows are pulled into LDS by the Tensor Data Mover in
// GATHER mode (4 descriptors x 16 indexed rows, 16-bit indices,
// tile_dim0=256 f32, pad 1 dword per 256 -> LDS row stride 257 dwords =>
// conflict-free bank access in the score loop), synchronized with
// s_wait_tensorcnt + workgroup barrier. Fallback: coalesced manual copy.
// score(m,h) = SCALE * sum_d [ (q+u)_c * K[j,c] + (q+v)_c * e(m)_c ],
// c = h*32+d, e(m)_c = bl_c + sum_p pg[m,p]*Wl[c,p]  (uK folded into (q+u)·K)
// ---------------------------------------------------------------------------
__global__ void __launch_bounds__(256) attn_kernel(
    const float* __restrict__ pg,            // (B,N,N,POS)
    const unsigned char* __restrict__ mask,  // (B,N)
    const float* __restrict__ Qb,            // (B*N, C)
    const float* __restrict__ Kb,            // (B*N, C)
    const float* __restrict__ Vb,            // (B*N, C)
    const float* __restrict__ Wl,            // (C, POS)
    const float* __restrict__ bl,            // (C)
    const float* __restrict__ u,             // (H*D) == C
    const float* __restrict__ v,             // (H*D) == C
    const int* __restrict__ nbhd_idx,        // (B*N, M)
    float* __restrict__ ctx)                 // (B*N, C)
{
  __shared__ float ks[M_ * KST_];          // 64 K rows, stride 257 (64.3 KB)
  __shared__ float qku[C_], qpv[C_];
  __shared__ float wl_s[C_ * POS_], bl_s[C_];
  __shared__ float pgs[M_ * POS_];
  __shared__ int   idx_s[M_];
  __shared__ unsigned char msk_s[M_];
  __shared__ float As[M_ * H_];

  const int row = blockIdx.x;
  const int b   = row / N_;
  const int t   = threadIdx.x;     // channel id 0..255

  if (t < M_) {
    int j = nbhd_idx[row * M_ + t];
    idx_s[t] = j;
    msk_s[t] = mask[b * N_ + j];
  }
  {
    float q  = Qb[(size_t)row * C_ + t];
    qku[t]   = q + u[t];           // (H,D) flattens to C exactly as c=h*32+d
    qpv[t]   = q + v[t];
    bl_s[t]  = bl[t];
  }
  for (int i = t; i < C_ * POS_; i += 256) wl_s[i] = Wl[i];
  __syncthreads();                 // idx_s visible to all waves / TDM issuer

#if HAVE_TDM
  // --- Tensor Data Mover: gather 64 indexed K rows into LDS --------------
  if (t < 32) {
    const unsigned long long gaddr =
        (unsigned long long)(uintptr_t)(Kb + (size_t)b * N_ * C_);
    for (int chunk = 0; chunk < 4; ++chunk) {
      tdm_u32x4 g0;
      // count=1 | gather_index_size=0 (16-bit) | gather_mode=1
      g0[0] = 1u | (1u << 31);
      g0[1] = (unsigned)(uintptr_t)&ks[chunk * 16 * KST_];  // LDS byte offset
      g0[2] = (unsigned)(gaddr & 0xFFFFFFFFull);
      g0[3] = (unsigned)((gaddr >> 32) & 0x01FFFFFFull) | (2u << 30); // type=2
      tdm_i32x8 g1;
      // data_size=2 (4B) | pad_enable | pad_interval=7 (256 dw) | pad_amount=0 (1 dw)
      g1[0] = (2 << 16) | (1 << 20) | (7 << 22);
      g1[1] = (int)((unsigned)(C_ & 0xFFFF) << 16);   // tensor_dim0[15:0]=256
      g1[2] = (int)((unsigned)(N_ & 0xFFFF) << 16);   // tensor_dim1[15:0]=2048
      g1[3] = (int)((unsigned)C_ << 16);              // tile_dim0=256
      g1[4] = 16;                                     // tile_dim1 = #indices
      g1[5] = C_;                                     // tensor_dim0_stride=256
      g1[6] = 0;
      g1[7] = 0;
      tdm_i32x4 g2, g3;
#pragma unroll
      for (int i = 0; i < 4; ++i) {
        g2[i] = idx_s[chunk * 16 + 2 * i]     | (idx_s[chunk * 16 + 2 * i + 1] << 16);
        g3[i] = idx_s[chunk * 16 + 8 + 2 * i] | (idx_s[chunk * 16 + 9 + 2 * i] << 16);
      }
#if __clang_major__ >= 23
      tdm_i32x8 gz = {};
      __builtin_amdgcn_tensor_load_to_lds(g0, g1, g2, g3, gz, 0);
#else
      __builtin_amdgcn_tensor_load_to_lds(g0, g1, g2, g3, 0);
#endif
    }
    __builtin_amdgcn_s_wait_tensorcnt(0);
  }
#else
  // --- Fallback: coalesced manual gather (one full row per 256 threads) --
  for (int i = t; i < M_ * C_; i += 256) {
    int m = i >> 8, c = i & 255;
    ks[m * KST_ + c] = Kb[((size_t)b * N_ + idx_s[m]) * C_ + c];
  }
#endif

  for (int i = t; i < M_ * POS_; i += 256) {
    int m = i / POS_, p = i % POS_;
    pgs[i] = pg[((size_t)row * N_ + idx_s[m]) * POS_ + p];
  }
  __syncthreads();

  // 512 (m,h) score pairs; lane<->m mapping (m = pair&63) makes the wave read
  // 32 consecutive padded LDS rows -> 32 distinct banks; wl/qku/qpv broadcast.
  for (int pair = t; pair < M_ * H_; pair += 256) {
    const int m = pair & 63, h = pair >> 6;
    const float* krow = ks + m * KST_;
    const float* pgm  = pgs + m * POS_;
    float s = 0.f;
#pragma unroll 4
    for (int d = 0; d < D_; ++d) {
      const int c = h * D_ + d;
      float e = bl_s[c];
#pragma unroll
      for (int p = 0; p < POS_; ++p) e += pgm[p] * wl_s[c * POS_ + p];
      s += qku[c] * krow[c] + qpv[c] * e;
    }
    s *= SCALE_;
    if (!msk_s[m]) s = -1e38f;
    As[m * H_ + h] = s;
  }
  __syncthreads();

  // masked softmax over M per head (8 heads -> 8 threads)
  if (t < H_) {
    float mx = -FLT_MAX;
    for (int m = 0; m < M_; ++m) mx = fmaxf(mx, As[m * H_ + t]);
    float sum = 0.f;
    for (int m = 0; m < M_; ++m) {
      float e = expf(As[m * H_ + t] - mx);
      As[m * H_ + t] = e;
      sum += e;
    }
    float inv = 1.f / sum;
    for (int m = 0; m < M_; ++m) As[m * H_ + t] *= inv;
  }
  __syncthreads();

  // weighted sum of gathered V rows (whole block reads one V row per m,
  // fully coalesced; V is 4MB -> L2 resident)
  {
    const int h = t >> 5;
    float acc = 0.f;
    for (int m = 0; m < M_; ++m)
      acc += As[m * H_ + h] * Vb[((size_t)b * N_ + idx_s[m]) * C_ + t];
    ctx[(size_t)row * C_ + t] = acc;
  }
}

// ---------------------------------------------------------------------------
// Host-side orchestration
// ---------------------------------------------------------------------------
extern "C" void kernel_launch(void* const* d_in, const int* in_sizes, int n_in,
                              void* d_out, int out_size, void* d_ws, size_t ws_size,
                              hipStream_t stream) {
  const float*         pg   = (const float*)d_in[0];          // (B,N,N,POS)
  const float*         cf   = (const float*)d_in[1];          // (B,N,C)
  const unsigned char* mask = (const unsigned char*)d_in[2];  // (B,N) bool
  const float* Wq = (const float*)d_in[3];
  const float* bq = (const float*)d_in[4];
  const float* Wk = (const float*)d_in[5];
  const float* bk = (const float*)d_in[6];
  const float* Wl = (const float*)d_in[7];
  const float* bl = (const float*)d_in[8];
  const float* u  = (const float*)d_in[9];
  const float* v  = (const float*)d_in[10];
  const float* Wi = (const float*)d_in[11];
  const float* bi = (const float*)d_in[12];
  const float* Wo = (const float*)d_in[13];
  const float* bo = (const float*)d_in[14];

  // workspace layout
  char*  ws   = (char*)d_ws;
  int*   nbhd = (int*)ws;                                   // 1 MB
  float* Qb   = (float*)(ws + (size_t)(1 << 20));           // 4 MB
  float* Kb   = Qb + (size_t)ROWS_ * C_;                    // 4 MB
  float* Vb   = Kb + (size_t)ROWS_ * C_;                    // 4 MB
  float* ctx  = Vb + (size_t)ROWS_ * C_;                    // 4 MB
  float* out  = (float*)d_out;

  // 1) neighborhoods (non-temporal single pass over the 201MB pairwise_g)
  topk_kernel<<<ROWS_, 256, 0, stream>>>(pg, mask, nbhd);

  // 2) fp32 WMMA projections: Q, K, V (project BEFORE gathering -> 64x fewer FLOPs)
  gemm_xwt_kernel<<<ROWS_ / 8, 256, 0, stream>>>(cf, Wq, bq, Qb, ROWS_);
  gemm_xwt_kernel<<<ROWS_ / 8, 256, 0, stream>>>(cf, Wk, bk, Kb, ROWS_);
  gemm_xwt_kernel<<<ROWS_ / 8, 256, 0, stream>>>(cf, Wi, bi, Vb, ROWS_);

  // 3) fused neighborhood attention (TDM gather of K rows into LDS)
  attn_kernel<<<ROWS_, 256, 0, stream>>>(pg, mask, Qb, Kb, Vb, Wl, bl, u, v,
                                         nbhd, ctx);

  // 4) output projection (fp32 WMMA) straight into d_out
  gemm_xwt_kernel<<<ROWS_ / 8, 256, 0, stream>>>(ctx, Wo, bo, out, ROWS_);
}